// LowPrecLinear_23175643529617
// MI455X (gfx1250) — compile-verified
//
#include <hip/hip_runtime.h>

typedef __attribute__((ext_vector_type(2))) float v2f;
typedef __attribute__((ext_vector_type(4))) float v4f;
typedef __attribute__((ext_vector_type(8))) float v8f;

namespace {
constexpr int M = 8192;
constexpr int K = 4096;
constexpr int N = 4096;

constexpr int BM   = 128;   // block tile M
constexpr int BN   = 128;   // block tile N
constexpr int KC   = 32;    // LDS K-chunk
constexpr int NCH  = K / KC;     // 128 chunks
constexpr int ST   = KC + 4;     // 36: row stride (floats) for both tiles -> conflict-free b64 frag reads
}

// Round-to-nearest-even fp32 mantissa to 10 bits (shift = 23-10 = 13).
// Bit-exact vs reference: rb = ((b>>13)&1) + (2^12 - 1); b = (b+rb) & ~(2^13-1)
__device__ __forceinline__ float round_m10(float f) {
  int b  = __float_as_int(f);
  int rb = ((b >> 13) & 1) + 0x0FFF;
  b = (b + rb) & (int)0xFFFFE000;
  return __int_as_float(b);
}

__global__ __launch_bounds__(256)
void lowprec_linear_wmma_f32(const float* __restrict__ x,      // [M,K]
                             const float* __restrict__ w,      // [N,K]
                             const float* __restrict__ bias,   // [N]
                             float* __restrict__ out)          // [M,N]
{
  // Both tiles row-major in their own dimension, K contiguous:
  //   As[m][k]  m in 0..127, k in 0..31   (A fragment: a.x=A[m][k], a.y=A[m][k+1] -> b64)
  //   Bs[n][k]  n in 0..127, k in 0..31   (B fragment: b.x=B^T[n][k]=B[k][n], b.y=B[k+1][n] -> b64)
  __shared__ float As[BM * ST];
  __shared__ float Bs[BN * ST];

  const int tid    = threadIdx.x;
  const int lane   = tid & 31;
  const int wave   = tid >> 5;
  const int lane16 = lane & 15;
  const int half   = lane >> 4;        // 0: lanes 0-15 (K 0,1) / 1: lanes 16-31 (K 2,3)

  const int bm0 = blockIdx.y * BM;
  const int bn0 = blockIdx.x * BN;

  // 8 waves arranged 4(M) x 2(N); each wave owns a 32x64 sub-tile = 2x4 WMMA tiles
  const int wm0 = (wave & 3) * 32;
  const int wn0 = (wave >> 2) * 64;

  // tile-loader mapping: 8 float4 columns x 32 rows, 4 row-steps of 32
  const int lc = tid & 7;    // float4 column within 32-wide K chunk
  const int lr = tid >> 3;   // row 0..31

  const float* xp = x + (size_t)(bm0 + lr) * K + lc * 4;
  const float* wp = w + (size_t)(bn0 + lr) * K + lc * 4;

  v8f acc[2][4] = {};

  // register double-buffer for the incoming chunk
  v4f ra[4], rb[4];
#pragma unroll
  for (int i = 0; i < 4; ++i) {
    ra[i] = *(const v4f*)(xp + (size_t)(32 * i) * K);
    rb[i] = *(const v4f*)(wp + (size_t)(32 * i) * K);
  }

  for (int ch = 0; ch < NCH; ++ch) {
    // ---- commit staged chunk to LDS ----
#pragma unroll
    for (int i = 0; i < 4; ++i) {
      *(v4f*)(&As[(lr + 32 * i) * ST + lc * 4]) = ra[i];
      *(v4f*)(&Bs[(lr + 32 * i) * ST + lc * 4]) = rb[i];
    }
    __syncthreads();

    // ---- issue next chunk's global loads; they fly while we do 64 WMMAs ----
    if (ch + 1 < NCH) {
      const int k1 = (ch + 1) * KC;
#pragma unroll
      for (int i = 0; i < 4; ++i) {
        ra[i] = *(const v4f*)(xp + (size_t)(32 * i) * K + k1);
        rb[i] = *(const v4f*)(wp + (size_t)(32 * i) * K + k1);
      }
      if (ch + 2 < NCH) {  // warm L2 one chunk further ahead
        __builtin_prefetch(xp + k1 + KC, 0, 0);
        __builtin_prefetch(wp + k1 + KC, 0, 0);
      }
    }

    // ---- compute: 8 K-steps of 4, 8 WMMA tiles each ----
#pragma unroll
    for (int kk = 0; kk < KC; kk += 4) {
      v2f a[2];
      v2f b[4];
#pragma unroll
      for (int mi = 0; mi < 2; ++mi)
        a[mi] = *(const v2f*)(&As[(wm0 + mi * 16 + lane16) * ST + kk + 2 * half]);
#pragma unroll
      for (int ni = 0; ni < 4; ++ni)
        b[ni] = *(const v2f*)(&Bs[(wn0 + ni * 16 + lane16) * ST + kk + 2 * half]);
#pragma unroll
      for (int mi = 0; mi < 2; ++mi)
#pragma unroll
        for (int ni = 0; ni < 4; ++ni)
          acc[mi][ni] = __builtin_amdgcn_wmma_f32_16x16x4_f32(
              false, a[mi], false, b[ni], (short)0, acc[mi][ni], false, false);
    }
    __syncthreads();

    // ---- reference semantics: round accumulator after every 128-deep K block (4 chunks) ----
    if ((ch & 3) == 3) {
#pragma unroll
      for (int mi = 0; mi < 2; ++mi)
#pragma unroll
        for (int ni = 0; ni < 4; ++ni)
#pragma unroll
          for (int j = 0; j < 8; ++j)
            acc[mi][ni][j] = round_m10(acc[mi][ni][j]);
    }
  }

  // ---- epilogue: out = round_m10(acc + bias) ----
#pragma unroll
  for (int ni = 0; ni < 4; ++ni) {
    const int col = bn0 + wn0 + ni * 16 + lane16;
    const float bv = bias[col];
#pragma unroll
    for (int mi = 0; mi < 2; ++mi) {
      const int rbase = bm0 + wm0 + mi * 16 + half * 8;  // C/D layout: vgpr j -> row j / j+8
#pragma unroll
      for (int j = 0; j < 8; ++j) {
        out[(size_t)(rbase + j) * N + col] = round_m10(acc[mi][ni][j] + bv);
      }
    }
  }
}

extern "C" void kernel_launch(void* const* d_in, const int* in_sizes, int n_in,
                              void* d_out, int out_size, void* d_ws, size_t ws_size,
                              hipStream_t stream) {
  (void)in_sizes; (void)n_in; (void)d_ws; (void)ws_size; (void)out_size;
  const float* x    = (const float*)d_in[0];
  const float* w    = (const float*)d_in[1];
  const float* bias = (const float*)d_in[2];
  float* out        = (float*)d_out;

  dim3 grid(N / BN, M / BM);   // (32, 64)
  dim3 block(256);
  lowprec_linear_wmma_f32<<<grid, block, 0, stream>>>(x, w, bias, out);
}